// IMUKalmanFilter_15556371546570
// MI455X (gfx1250) — compile-verified
//
#include <hip/hip_runtime.h>
#include <hip/hip_bf16.h>
#include <math.h>

typedef float v2f __attribute__((ext_vector_type(2)));
typedef float v8f __attribute__((ext_vector_type(8)));

// ---------------- small 3x3 helpers (run redundantly on every lane) ----------------
__device__ inline void skew3f(const float v[3], float M[9]) {
    M[0] = 0.f;   M[1] = -v[2]; M[2] = v[1];
    M[3] = v[2];  M[4] = 0.f;   M[5] = -v[0];
    M[6] = -v[1]; M[7] = v[0];  M[8] = 0.f;
}
__device__ inline void m3mul(const float A[9], const float B[9], float C[9]) {
    for (int i = 0; i < 3; ++i)
        for (int j = 0; j < 3; ++j)
            C[i*3+j] = A[i*3+0]*B[0*3+j] + A[i*3+1]*B[1*3+j] + A[i*3+2]*B[2*3+j];
}
__device__ inline void m3T(const float A[9], float At[9]) {
    for (int i = 0; i < 3; ++i)
        for (int j = 0; j < 3; ++j)
            At[i*3+j] = A[j*3+i];
}
__device__ inline void m3v(const float A[9], const float v[3], float r[3]) {
    for (int i = 0; i < 3; ++i)
        r[i] = A[i*3+0]*v[0] + A[i*3+1]*v[1] + A[i*3+2]*v[2];
}
__device__ inline void m3Tv(const float A[9], const float v[3], float r[3]) {
    for (int i = 0; i < 3; ++i)
        r[i] = A[0*3+i]*v[0] + A[1*3+i]*v[1] + A[2*3+i]*v[2];
}
__device__ inline void exp_so3(const float phi[3], float C[9]) {
    float n2 = phi[0]*phi[0] + phi[1]*phi[1] + phi[2]*phi[2];
    float n  = sqrtf(n2);
    float pk[9], pk2[9];
    skew3f(phi, pk);
    m3mul(pk, pk, pk2);
    float ca, cb;
    if (n < 1e-8f) { ca = 1.0f; cb = 0.5f; }
    else           { ca = sinf(n)/n; cb = (1.0f - cosf(n))/n2; }
    for (int i = 0; i < 9; ++i) {
        float d = (i == 0 || i == 4 || i == 8) ? 1.0f : 0.0f;
        C[i] = d + ca*pk[i] + cb*pk2[i];
    }
}
__device__ inline void log_so3(const float C[9], float r[3]) {
    float tr = C[0] + C[4] + C[8];
    float cv = fminf(fmaxf(0.5f*(tr - 1.0f), -1.0f), 1.0f);
    float pn = acosf(cv);
    float s  = sinf(pn);
    float v0 = C[7]-C[5], v1 = C[2]-C[6], v2 = C[3]-C[1];
    float f  = (s > 1e-6f) ? (pn / (2.0f*s)) : 0.5f;
    r[0] = f*v0; r[1] = f*v1; r[2] = f*v2;
}
__device__ inline void jlinv(const float phi[3], float J[9]) {
    float n2 = phi[0]*phi[0] + phi[1]*phi[1] + phi[2]*phi[2];
    float n  = sqrtf(n2);
    if (n <= 1e-6f) {
        float pk[9]; skew3f(phi, pk);
        for (int i = 0; i < 9; ++i) {
            float d = (i == 0 || i == 4 || i == 8) ? 1.0f : 0.0f;
            J[i] = d - 0.5f*pk[i];
        }
    } else {
        float a[3] = { phi[0]/n, phi[1]/n, phi[2]/n };
        float hn = 0.5f*n;
        float cot = cosf(hn)/sinf(hn);
        float h = hn*cot;
        float ask[9]; skew3f(a, ask);
        for (int i = 0; i < 3; ++i)
            for (int j = 0; j < 3; ++j) {
                float d = (i == j) ? 1.0f : 0.0f;
                J[i*3+j] = h*d + (1.0f-h)*a[i]*a[j] - hn*ask[i*3+j];
            }
    }
}
__device__ inline void inv6(const float S[36], float Si[36]) {
    float A[6][12];
    for (int i = 0; i < 6; ++i)
        for (int j = 0; j < 6; ++j) {
            A[i][j]   = S[i*6+j];
            A[i][6+j] = (i == j) ? 1.0f : 0.0f;
        }
    for (int k = 0; k < 6; ++k) {
        float ip = 1.0f / A[k][k];
        for (int j = 0; j < 12; ++j) A[k][j] *= ip;
        for (int i = 0; i < 6; ++i) {
            if (i == k) continue;
            float f = A[i][k];
            for (int j = 0; j < 12; ++j) A[i][j] -= f*A[k][j];
        }
    }
    for (int i = 0; i < 6; ++i)
        for (int j = 0; j < 6; ++j)
            Si[i*6+j] = A[i][6+j];
}

// ---------------- wave-cooperative 32x32 f32 matmul via V_WMMA_F32_16X16X4_F32 ----
// Cd = Am * Bm (TB=false) or Am * Bm^T (TB=true). All 32x32 row-major in LDS,
// K limited to 4*kchunks (zero padding beyond the live 18 (or 12) columns).
template<bool TB>
__device__ inline void wave_mm(float (*Cd)[32], const float (*Am)[32],
                               const float (*Bm)[32], int kchunks) {
    __syncthreads();
    const int lane = threadIdx.x;
    const int half = lane >> 4;   // 0: lanes 0-15, 1: lanes 16-31
    const int l15  = lane & 15;
    for (int ti = 0; ti < 2; ++ti) {
        for (int tj = 0; tj < 2; ++tj) {
            v8f acc = {0.f,0.f,0.f,0.f,0.f,0.f,0.f,0.f};
            const int m = ti*16 + l15;
            const int n = tj*16 + l15;
            for (int kc = 0; kc < kchunks; ++kc) {
                const int k0 = kc*4 + half*2;
                v2f av; av.x = Am[m][k0]; av.y = Am[m][k0+1];
                v2f bv;
                if (TB) { bv.x = Bm[n][k0];   bv.y = Bm[n][k0+1]; }
                else    { bv.x = Bm[k0][n];   bv.y = Bm[k0+1][n]; }
                acc = __builtin_amdgcn_wmma_f32_16x16x4_f32(
                        false, av, false, bv, (short)0, acc, false, false);
            }
#pragma unroll
            for (int vv = 0; vv < 8; ++vv)
                Cd[ti*16 + half*8 + vv][tj*16 + l15] = acc[vv];
        }
    }
    __syncthreads();
}

// ---------------- main EKF kernel: one wave32 per batch chain ----------------
__global__ __launch_bounds__(32)
void ekf_vio_kernel(const float* __restrict__ imu, const float* __restrict__ Qimu,
                    const float* __restrict__ pose0, const float* __restrict__ state0,
                    const float* __restrict__ covar0, const float* __restrict__ vis,
                    const float* __restrict__ visR, const float* __restrict__ Tic,
                    float* __restrict__ out, int Bn, int Tn, int Nimu) {
    __shared__ float sP[32][32];    // covariance (padded 18x18)
    __shared__ float sPhi[32][32];  // Phi / later Q
    __shared__ float sA[32][32];    // F / PG / (I-KH) / U
    __shared__ float sB[32][32];    // FF / Pnew / est_covar
    __shared__ float sT[32][32];    // matmul temp
    __shared__ float sQ[32][32];    // Qimu padded (persistent)
    __shared__ float sK[18][6];     // Kalman gain

    const int b    = blockIdx.x;
    const int lane = threadIdx.x;

    // load Qimu (12x12) and initial covariance (18x18), zero-padded to 32x32
    for (int c = 0; c < 32; ++c) {
        float q = 0.f;
        if (lane < 12 && c < 12) q = Qimu[lane*12 + c];
        sQ[lane][c] = q;
        float p = 0.f;
        if (lane < 18 && c < 18) p = covar0[(size_t)b*324 + lane*18 + c];
        sP[lane][c] = p;
    }
    __syncthreads();

    // per-lane carried state (identical across lanes)
    float poseR[16], stR[24];
    for (int i = 0; i < 16; ++i) poseR[i] = pose0[(size_t)b*16 + i];
    for (int i = 0; i < 24; ++i) stR[i]   = state0[(size_t)b*24 + i];
    float Ccal[9], rcal[3];
    for (int i = 0; i < 3; ++i) {
        for (int j = 0; j < 3; ++j) Ccal[i*3+j] = Tic[(size_t)b*16 + i*4 + j];
        rcal[i] = Tic[(size_t)b*16 + i*4 + 3];
    }

    // output row 0 = initial pose/state/covar
    {
        size_t base = ((size_t)b*(Tn+1)) * 364;
        if (lane < 16) out[base + lane]      = poseR[lane];
        if (lane < 24) out[base + 16 + lane] = stR[lane];
        for (int e = lane; e < 324; e += 32) out[base + 40 + e] = sP[e/18][e%18];
    }

    for (int t = 0; t < Tn; ++t) {
        // ======================= PREDICT =======================
        const float* seq = imu + ((size_t)(b*Tn + t)) * Nimu * 7;
        float gk[3] = { stR[0], stR[1], stR[2] };
        float Ck[9]; for (int i = 0; i < 9; ++i) Ck[i] = stR[3+i];
        float rk[3] = { stR[12], stR[13], stR[14] };
        float vk[3] = { stR[15], stR[16], stR[17] };
        float bw[3] = { stR[18], stR[19], stR[20] };
        float ba[3] = { stR[21], stR[22], stR[23] };
        float t_acc = 0.f;
        float Cacc[9] = {1,0,0, 0,1,0, 0,0,1};
        float racc[3] = {0,0,0}, vacc[3] = {0,0,0};

        for (int j = 0; j < Nimu-1; ++j) {
            float dt = seq[(j+1)*7] - seq[j*7];
            float gy[3] = { seq[j*7+1], seq[j*7+2], seq[j*7+3] };
            float ac[3] = { seq[j*7+4], seq[j*7+5], seq[j*7+6] };
            float w[3]  = { gy[0]-bw[0], gy[1]-bw[1], gy[2]-bw[2] };
            float av[3] = { ac[0]-ba[0], ac[1]-ba[1], ac[2]-ba[2] };
            float wsk[9]; skew3f(w, wsk);
            float tmp3[3] = { vk[0]-gk[0]*t_acc+vacc[0],
                              vk[1]-gk[1]*t_acc+vacc[1],
                              vk[2]-gk[2]*t_acc+vacc[2] };
            float vv3[3]; m3Tv(Cacc, tmp3, vv3);
            float vsk[9]; skew3f(vv3, vsk);
            float phiw[3] = { dt*w[0], dt*w[1], dt*w[2] };
            float expw[9]; exp_so3(phiw, expw);

            // --- build F into sA (each lane owns row `lane`) ---
            {
                const int r = lane;
                float Cvsk[9]; m3mul(Cacc, vsk, Cvsk);
                float Ctg[3];  m3Tv(Cacc, gk, Ctg);
                float skCtg[9]; skew3f(Ctg, skCtg);
                for (int c = 0; c < 32; ++c) {
                    float val = 0.f;
                    if (r >= 3 && r < 6) {
                        int i = r - 3;
                        if (c >= 3 && c < 6)        val = -wsk[i*3 + (c-3)];
                        else if (c >= 12 && c < 15) val = (c-12 == i) ? -1.f : 0.f;
                    } else if (r >= 6 && r < 9) {
                        int i = r - 6;
                        if (c >= 3 && c < 6)        val = -Cvsk[i*3 + (c-3)];
                        else if (c >= 9 && c < 12)  val = Cacc[i*3 + (c-9)];
                    } else if (r >= 9 && r < 12) {
                        int i = r - 9;
                        if (c < 3)                  val = -Cacc[c*3 + i];     // -Cacc^T
                        else if (c >= 3 && c < 6)   val = -skCtg[i*3 + (c-3)];
                        else if (c >= 9 && c < 12)  val = -wsk[i*3 + (c-9)];
                        else if (c >= 12 && c < 15) val = -vsk[i*3 + (c-12)];
                        else if (c >= 15 && c < 18) val = (c-15 == i) ? -1.f : 0.f;
                    }
                    sA[r][c] = val;
                }
            }
            // FF = F @ F
            wave_mm<false>(sB, sA, sA, 5);
            // Phi = I + F dt + 0.5 FF dt^2, with patches
            {
                const int r = lane;
                const float dt2h = 0.5f*dt*dt;
                for (int c = 0; c < 32; ++c) {
                    float val = 0.f;
                    if (r < 18 && c < 18)
                        val = ((r == c) ? 1.f : 0.f) + dt*sA[r][c] + dt2h*sB[r][c];
                    sPhi[r][c] = val;
                }
                if (r >= 6 && r < 9) { sPhi[r][12] = 0.f; sPhi[r][13] = 0.f; sPhi[r][14] = 0.f; }
                if (r >= 3 && r < 6) { int i = r-3; for (int c = 0; c < 3; ++c) sPhi[r][3+c] = expw[c*3+i]; }
                if (r >= 9 && r < 12){ int i = r-9; for (int c = 0; c < 3; ++c) sPhi[r][9+c] = expw[c*3+i]; }
            }
            // Pnew = Phi @ P @ Phi^T
            wave_mm<false>(sT, sPhi, sP, 5);
            wave_mm<true>(sB, sT, sPhi, 5);
            // PG = Phi @ G (G sparse) into sA
            {
                const int r = lane;
                for (int c = 0; c < 32; ++c) sA[r][c] = 0.f;
                if (r < 18) {
                    for (int c = 0; c < 3; ++c) {
                        sA[r][c]   = -sPhi[r][3+c]
                                     - (sPhi[r][9]*vsk[c] + sPhi[r][10]*vsk[3+c] + sPhi[r][11]*vsk[6+c]);
                        sA[r][3+c] =  sPhi[r][12+c];
                        sA[r][6+c] = -sPhi[r][9+c];
                        sA[r][9+c] =  sPhi[r][15+c];
                    }
                }
            }
            // Q = (PG @ Qimu) @ PG^T
            wave_mm<false>(sT, sA, sQ, 3);
            wave_mm<true>(sPhi, sT, sA, 3);
            // P = force_sym(Pnew + dt*Q)  (upper triangle wins)
            {
                const int r = lane;
                for (int c = 0; c < 32; ++c) {
                    float val = 0.f;
                    if (r < 18 && c < 18) {
                        int i = (r <= c) ? r : c, jj = (r <= c) ? c : r;
                        val = sB[i][jj] + dt*sPhi[i][jj];
                    }
                    sP[r][c] = val;
                }
            }
            __syncthreads();
            // scalar accumulators
            float Ca[3]; m3v(Cacc, av, Ca);
            for (int i = 0; i < 3; ++i) {
                racc[i] += vacc[i]*dt + 0.5f*dt*dt*Ca[i];
                vacc[i] += dt*Ca[i];
            }
            float Cn[9]; m3mul(Cacc, expw, Cn);
            for (int i = 0; i < 9; ++i) Cacc[i] = Cn[i];
            t_acc += dt;
        }

        // predicted state
        float Cp[9]; m3mul(Ck, Cacc, Cp);
        float tv[3] = { vk[0]-gk[0]*t_acc+vacc[0], vk[1]-gk[1]*t_acc+vacc[1], vk[2]-gk[2]*t_acc+vacc[2] };
        float vp[3]; m3Tv(Cacc, tv, vp);
        float rp[3];
        for (int i = 0; i < 3; ++i)
            rp[i] = rk[i] + vk[i]*t_acc - 0.5f*gk[i]*t_acc*t_acc + racc[i];

        // ======================= UPDATE =======================
        const float* vm = vis  + ((size_t)(b*Tn + t)) * 6;
        const float* Rm = visR + ((size_t)(b*Tn + t)) * 36;
        float vm0[3] = { vm[0], vm[1], vm[2] };
        float E[9];  exp_so3(vm0, E);
        float CcalT[9]; m3T(Ccal, CcalT);
        float CpT[9];   m3T(Cp, CpT);
        float T1[9], T2[9], T3[9];
        m3mul(E, CcalT, T1); m3mul(T1, CpT, T2); m3mul(T2, Ccal, T3);
        float resr[3]; log_so3(T3, resr);
        float Cprc[3]; m3v(Cp, rcal, Cprc);
        float d1[3]; m3Tv(Ccal, Cprc, d1);
        float rpr[3] = { rp[0]-rcal[0], rp[1]-rcal[1], rp[2]-rcal[2] };
        float d2[3]; m3Tv(Ccal, rpr, d2);
        float resid[6] = { resr[0], resr[1], resr[2],
                           vm[3]-d1[0]-d2[0], vm[4]-d1[1]-d2[1], vm[5]-d1[2]-d2[2] };
        // final H (after H = -H): H[0:3,3:6]=Jlinv(-resr)@Ccal^T@Cp ; H[3:6,3:6]=-(Ccal^T@Cp@skew(rcal)) ; H[3:6,6:9]=Ccal^T
        float nresr[3] = { -resr[0], -resr[1], -resr[2] };
        float Jl[9]; jlinv(nresr, Jl);
        float CtC[9]; m3mul(CcalT, Cp, CtC);
        float H0[9];  m3mul(Jl, CtC, H0);
        float skrc[9]; skew3f(rcal, skrc);
        float H1m[9]; m3mul(CtC, skrc, H1m);
        float Hc[36];   // H columns 3..8 only
        for (int i = 0; i < 3; ++i)
            for (int jj = 0; jj < 6; ++jj) {
                Hc[i*6+jj]     = (jj < 3) ?  H0[i*3+jj]        : 0.f;
                Hc[(3+i)*6+jj] = (jj < 3) ? -H1m[i*3+jj]       : CcalT[i*3+(jj-3)];
            }
        // S = Hc @ P[3:9,3:9] @ Hc^T + R ; Sinv (redundant per-lane)
        float HPc[36];
        for (int i = 0; i < 6; ++i)
            for (int kk = 0; kk < 6; ++kk) {
                float a = 0.f;
                for (int l = 0; l < 6; ++l) a += Hc[i*6+l] * sP[3+l][3+kk];
                HPc[i*6+kk] = a;
            }
        float Smat[36];
        for (int i = 0; i < 6; ++i)
            for (int jj = 0; jj < 6; ++jj) {
                float a = 0.f;
                for (int k = 0; k < 6; ++k) a += HPc[i*6+k] * Hc[jj*6+k];
                Smat[i*6+jj] = a + Rm[i*6+jj];
            }
        float Sinv[36]; inv6(Smat, Sinv);
        // K = P H^T Sinv (distributed into LDS)
        for (int e = lane; e < 108; e += 32) {
            int r = e/6, jj = e%6;
            float acc = 0.f;
            for (int m = 0; m < 6; ++m) {
                float pht = 0.f;
                for (int k = 0; k < 6; ++k) pht += sP[r][3+k] * Hc[m*6+k];
                acc += pht * Sinv[m*6+jj];
            }
            sK[r][jj] = acc;
        }
        __syncthreads();
        float err[18];
        for (int r = 0; r < 18; ++r) {
            float a = 0.f;
            for (int jj = 0; jj < 6; ++jj) a += sK[r][jj] * resid[jj];
            err[r] = a;
        }
        // A = I - K@H into sA, est_covar = A @ P (WMMA)
        {
            const int r = lane;
            for (int c = 0; c < 32; ++c) {
                float val = 0.f;
                if (r < 18 && c < 18) {
                    val = (r == c) ? 1.f : 0.f;
                    if (c >= 3 && c < 9) {
                        float kh = 0.f;
                        for (int jj = 0; jj < 6; ++jj) kh += sK[r][jj] * Hc[jj*6 + (c-3)];
                        val -= kh;
                    }
                }
                sA[r][c] = val;
            }
        }
        wave_mm<false>(sB, sA, sP, 5);
        // est_state
        float ge[3] = { gk[0]+err[0], gk[1]+err[1], gk[2]+err[2] };
        float ephi[3] = { err[3], err[4], err[5] };
        float Ee[9]; exp_so3(ephi, Ee);
        float Ce[9]; m3mul(Cp, Ee, Ce);
        float re[3]  = { rp[0]+err[6],  rp[1]+err[7],  rp[2]+err[8]  };
        float ve[3]  = { vp[0]+err[9],  vp[1]+err[10], vp[2]+err[11] };
        float bwe[3] = { bw[0]+err[12], bw[1]+err[13], bw[2]+err[14] };
        float bae[3] = { ba[0]+err[15], ba[1]+err[16], ba[2]+err[17] };

        // ======================= COMPOSITION =======================
        float Cte[9]; m3T(Ce, Cte);
        float NP[16];
        for (int i = 0; i < 3; ++i)
            for (int jj = 0; jj < 3; ++jj)
                NP[i*4+jj] = Cte[i*3+0]*poseR[0*4+jj] + Cte[i*3+1]*poseR[1*4+jj] + Cte[i*3+2]*poseR[2*4+jj];
        float pr[3] = { poseR[3]-re[0], poseR[7]-re[1], poseR[11]-re[2] };
        float Ctp[3]; m3v(Cte, pr, Ctp);
        NP[3] = Ctp[0]; NP[7] = Ctp[1]; NP[11] = Ctp[2];
        NP[12] = 0.f; NP[13] = 0.f; NP[14] = 0.f; NP[15] = 1.f;
        float ng[3]; m3v(Cte, ge, ng);
        float skng[9]; skew3f(ng, skng);
        // U into sA ; new_covar = force_sym(U @ ec @ U^T)
        {
            const int r = lane;
            for (int c = 0; c < 32; ++c) {
                float val = 0.f;
                if (r < 3) {
                    if (c < 3)      val = Cte[r*3+c];
                    else if (c < 6) val = skng[r*3 + (c-3)];
                } else if (r >= 9 && r < 18) {
                    val = (c == r) ? 1.f : 0.f;
                }
                sA[r][c] = val;
            }
        }
        wave_mm<false>(sT, sA, sB, 5);
        wave_mm<true>(sPhi, sT, sA, 5);
        {
            const int r = lane;
            for (int c = 0; c < 32; ++c) {
                float val = 0.f;
                if (r < 18 && c < 18) {
                    int i = (r <= c) ? r : c, jj = (r <= c) ? c : r;
                    val = sPhi[i][jj];
                }
                sP[r][c] = val;
            }
        }
        __syncthreads();

        // write outputs row t+1
        {
            size_t base = ((size_t)b*(Tn+1) + (t+1)) * 364;
            if (lane < 16) out[base + lane] = NP[lane];
            float ns[24];
            ns[0] = ng[0]; ns[1] = ng[1]; ns[2] = ng[2];
            ns[3] = 1.f; ns[4] = 0.f; ns[5] = 0.f;
            ns[6] = 0.f; ns[7] = 1.f; ns[8] = 0.f;
            ns[9] = 0.f; ns[10] = 0.f; ns[11] = 1.f;
            ns[12] = 0.f; ns[13] = 0.f; ns[14] = 0.f;
            ns[15] = ve[0];  ns[16] = ve[1];  ns[17] = ve[2];
            ns[18] = bwe[0]; ns[19] = bwe[1]; ns[20] = bwe[2];
            ns[21] = bae[0]; ns[22] = bae[1]; ns[23] = bae[2];
            if (lane < 24) out[base + 16 + lane] = ns[lane];
            for (int e = lane; e < 324; e += 32) out[base + 40 + e] = sP[e/18][e%18];
            // carry state
            for (int i = 0; i < 16; ++i) poseR[i] = NP[i];
            for (int i = 0; i < 24; ++i) stR[i] = ns[i];
        }
        __syncthreads();
    }
}

extern "C" void kernel_launch(void* const* d_in, const int* in_sizes, int n_in,
                              void* d_out, int out_size, void* d_ws, size_t ws_size,
                              hipStream_t stream) {
    (void)n_in; (void)out_size; (void)d_ws; (void)ws_size;
    const float* imu    = (const float*)d_in[1];
    const float* Qimu   = (const float*)d_in[2];
    const float* pose0  = (const float*)d_in[3];
    const float* state0 = (const float*)d_in[4];
    const float* covar0 = (const float*)d_in[5];
    const float* vis    = (const float*)d_in[6];
    const float* visR   = (const float*)d_in[7];
    const float* Tic    = (const float*)d_in[8];
    float* out = (float*)d_out;

    int Bn   = in_sizes[3] / 16;
    int Tn   = (Bn > 0) ? in_sizes[6] / (6 * Bn) : 1;
    int Nimu = (Bn > 0 && Tn > 0) ? in_sizes[1] / (Bn * Tn * 7) : 2;

    hipLaunchKernelGGL(ekf_vio_kernel, dim3(Bn), dim3(32), 0, stream,
                       imu, Qimu, pose0, state0, covar0, vis, visR, Tic,
                       out, Bn, Tn, Nimu);
}